// ParticleTrackingLayer_22943715295733
// MI455X (gfx1250) — compile-verified
//
#include <hip/hip_runtime.h>

// Particle advection via bilinear interpolation on a periodic 512x512 field.
// Memory-bound gather kernel: ~128 MB of HBM traffic, ~130 MFLOP total
// -> decisively bandwidth-bound; no matrix structure, so the optimized path
// on MI455X is the vmem path: wave32, 8-byte vector loads for (x,y)/(u,v)
// pairs, scalar (SGPR) batch base + 32-bit per-lane offsets for the gathers,
// and non-temporal hints on the streamed input/output so the velocity field
// stays resident in the 192 MB L2 for the random corner gathers.

typedef float v2f __attribute__((ext_vector_type(2)));

#define P_B    32
#define P_NX   512
#define P_NY   512
#define P_NSEN 131072
#define P_N    (P_B * P_NSEN)          // 4,194,304 particles
#define TWO_PI_D 6.283185307179586476925286766559

__global__ __launch_bounds__(256) void
ParticleTrackingLayer_22943715295733_kernel(const float* __restrict__ xpyp,
                                            const float* __restrict__ ufvf,
                                            float* __restrict__ out)
{
    const int p = blockIdx.x * blockDim.x + threadIdx.x;
    if (p >= P_N) return;

    // p/NSEN is identical for every lane of the block (NSEN % 256 == 0):
    // force it into an SGPR so the field base address is scalar and the
    // gathers use saddr + 32-bit voffset form.
    const int b = __builtin_amdgcn_readfirstlane((int)((unsigned)p >> 17));

    // (x, y) pair is contiguous -> single 8-byte non-temporal load (stream).
    const v2f xy = __builtin_nontemporal_load((const v2f*)xpyp + p);
    const float x = xy.x;
    const float y = xy.y;

    const float inv_dx = (float)(512.0 / TWO_PI_D);

    // ix1 = int32(x / dx); truncation toward zero, x >= 0, ix1 in [0, 511].
    const float xs = x * inv_dx;
    const float ys = y * inv_dx;
    const int ix1 = (int)xs;
    const int iy1 = (int)ys;

    // alpha = (x - x_grid[ix1]) / dx  ==  x/dx - ix1  (to rounding)
    const float alpha = xs - (float)ix1;
    const float beta  = ys - (float)iy1;

    // ix2 = ix1 + 1 - (ix1 // 511) * 512  -> periodic wrap at 511
    const int ix2 = (ix1 == P_NX - 1) ? 0 : ix1 + 1;
    const int iy2 = (iy1 == P_NY - 1) ? 0 : iy1 + 1;

    // Scalar batch base (SGPR); per-lane 32-bit element offsets.
    const v2f* __restrict__ f = (const v2f*)ufvf + (size_t)b * (P_NX * P_NY);
    const unsigned r1 = (unsigned)(ix1 * P_NY);
    const unsigned r2 = (unsigned)(ix2 * P_NY);
    const v2f f11 = f[r1 + (unsigned)iy1];
    const v2f f12 = f[r1 + (unsigned)iy2];
    const v2f f21 = f[r2 + (unsigned)iy1];
    const v2f f22 = f[r2 + (unsigned)iy2];

    const float omA = 1.0f - alpha;
    const float omB = 1.0f - beta;
    const float w11 = omA   * omB;
    const float w21 = alpha * omB;
    const float w22 = alpha * beta;
    const float w12 = omA   * beta;

    const float up = w11 * f11.x + w21 * f21.x + w22 * f22.x + w12 * f12.x;
    const float vp = w11 * f11.y + w21 * f21.y + w22 * f22.y + w12 * f12.y;

    const float dt = 0.01f;
    // Tuple output concatenated flat: [0, N) = x', [N, 2N) = y'.
    __builtin_nontemporal_store(x + up * dt, out + p);
    __builtin_nontemporal_store(y + vp * dt, out + P_N + p);
}

extern "C" void kernel_launch(void* const* d_in, const int* in_sizes, int n_in,
                              void* d_out, int out_size, void* d_ws, size_t ws_size,
                              hipStream_t stream)
{
    (void)in_sizes; (void)n_in; (void)out_size; (void)d_ws; (void)ws_size;
    const float* xpyp = (const float*)d_in[0];   // (B, NSEN, 2) f32
    const float* ufvf = (const float*)d_in[1];   // (B, NX, NY, 2) f32
    float* out = (float*)d_out;                  // (B*NSEN) x' then (B*NSEN) y'

    const int threads = 256;                     // 8 wave32s per block
    const int blocks  = (P_N + threads - 1) / threads;   // 16384
    ParticleTrackingLayer_22943715295733_kernel<<<blocks, threads, 0, stream>>>(
        xpyp, ufvf, out);
}